// LinearCaps_48060684042598
// MI455X (gfx1250) — compile-verified
//
#include <hip/hip_runtime.h>
#include <math.h>

// ---------------------------------------------------------------------------
// Capsule dynamic routing, MI455X (gfx1250, wave32).
//   u[B,c,D,d] = sum_C x[B,C,D] * w[c,C,D,d]   (never materialized in HBM)
//   3 routing iters fused into 3 passes; u recomputed per pass via
//   V_WMMA_F32_16X16X4_F32. w / x tiles streamed into LDS with the Tensor
//   Data Mover (tensor_load_to_lds + s_wait_tensorcnt), double-buffered.
// ---------------------------------------------------------------------------

#define B_TOT  64
#define C_IN   8
#define D_TOT  1152
#define C_OUT  16
#define D_OUT  64

#define BT     16                 // batch tile (WMMA M)
#define TD     32                 // D-values per workgroup
#define TDC    4                  // D-values per TDM chunk
#define NCH    (TD / TDC)         // chunks per workgroup (8)
#define NW     16                 // waves per block, wave w handles c = w
#define BLK    (NW * 32)          // 512 threads = 16 wave32
#define W_CSTRIDE (D_TOT * D_OUT) // stride of C index inside w (73728 floats)
#define WCHUNK (C_IN * TDC * D_OUT) // floats per (c, chunk) w tile = 2048

typedef float        v2f __attribute__((ext_vector_type(2)));
typedef float        v8f __attribute__((ext_vector_type(8)));
typedef unsigned int v4u __attribute__((ext_vector_type(4)));
typedef int          v8i __attribute__((ext_vector_type(8)));
typedef int          v4i __attribute__((ext_vector_type(4)));
typedef unsigned int u32;
typedef unsigned long long u64;

// Build a D# (ISA ch.8) for a 3D tile of 4-byte elements and issue
// TENSOR_LOAD_TO_LDS (6-arg clang-23 form: g0, g1, g2, g3, extra, cpol).
// dim0 is contiguous; stride0/stride1 (elements) step dim1/dim2.  lds_off is
// the workgroup-relative LDS byte offset (low 32 bits of the generic shared
// pointer, per the flat->LDS aperture rule).
__device__ __forceinline__ void tdm_load_3d(
    u32 lds_off, const float* gptr,
    u32 tile_d0, u32 tile_d1, u32 tile_d2,
    u32 tens_d0, u32 tens_d1, u32 tens_d2,
    u64 stride0, u64 stride1)
{
    const u64 ga = (u64)(uintptr_t)gptr;

    v4u g0;
    g0[0] = 1u;                                             // count=1 (valid)
    g0[1] = lds_off;                                        // lds_addr
    g0[2] = (u32)(ga & 0xFFFFFFFFu);                        // global_addr lo
    g0[3] = (u32)((ga >> 32) & 0x01FFFFFFu) | (2u << 30);   // addr hi | type=2

    u32 g1w[8];
    g1w[0] = (2u << 16);                                    // data_size = 4B
    g1w[1] = (tens_d0 & 0xFFFFu) << 16;                     // tensor_dim0 lo16
    g1w[2] = (tens_d0 >> 16) | ((tens_d1 & 0xFFFFu) << 16); // td0 hi | td1 lo
    g1w[3] = (tens_d1 >> 16) | ((tile_d0 & 0xFFFFu) << 16); // td1 hi | tile0
    g1w[4] = (tile_d1 & 0xFFFFu) | ((tile_d2 & 0xFFFFu) << 16);
    g1w[5] = (u32)(stride0 & 0xFFFFFFFFu);                  // dim0_stride lo32
    g1w[6] = (u32)((stride0 >> 32) & 0xFFFFu) |             // dim0_stride hi16
             ((u32)(stride1 & 0xFFFFu) << 16);              // dim1_stride lo16
    g1w[7] = (u32)((stride1 >> 16) & 0xFFFFFFFFu);          // dim1_stride hi32
    v8i g1;
    for (int i = 0; i < 8; ++i) g1[i] = (int)g1w[i];

    v4i g2;
    g2[0] = (int)tens_d2;   // tensor_dim2
    g2[1] = 0;              // tensor_dim3 (unused)
    g2[2] = 0;              // tensor_dim2_stride lo (no dim3)
    g2[3] = 0;              // stride hi | tile_dim3 = 0
    v4i g3 = (v4i)(0);
    v8i gx = (v8i)(0);      // extra group (clang-23 6-arg form), zero-filled

    __builtin_amdgcn_tensor_load_to_lds(g0, g1, g2, g3, gx, 0);
}

// MODE 0: iter1 s-pass, cc = 1/64 (softmax of zeros), no b, no v.
// MODE 1: fused iter1->2: b = sum_c u*v1 (b_old==0), write b, cc=softmax, s+=cc*u.
// MODE 2: fused iter2->3: read b, b += sum_c u*v2 (not stored), cc=softmax, s+=cc*u.
template <int MODE>
__global__ __launch_bounds__(BLK) void routing_pass(
    const float* __restrict__ x,     // (B, C, D)
    const float* __restrict__ w,     // (c, C, D, d)
    const float* __restrict__ v,     // (B, c, d)
    float* __restrict__ bglob,       // (B, D, d)
    float* __restrict__ sacc)        // (B, c, d) accumulated via atomics
{
    __shared__ float xs[BT * C_IN * TD];   // 16 KB x tile (TDM-filled)
    __shared__ float bsum[BT * D_OUT];     // 4 KB  sum_c u*v for current D
    __shared__ float ccs[BT * D_OUT];      // 4 KB  softmax coefficients
    extern __shared__ float wl[];          // 256 KB: [c][2 bufs][WCHUNK]

    const int tid  = threadIdx.x;
    const int lane = tid & 31;
    const int half = lane >> 4;
    const int l16  = lane & 15;
    const int wave = tid >> 5;
    const int c    = __builtin_amdgcn_readfirstlane(wave);  // SGPR c

    const int b0 = blockIdx.y * BT;
    const int D0 = blockIdx.x * TD;

    // ---- TDM: stage x[b0:b0+16, 0:8, D0:D0+TD] -> xs ([b][C][D]) ----
    if (c == 0) {
        tdm_load_3d((u32)(uintptr_t)&xs[0],
                    x + (size_t)b0 * C_IN * D_TOT + D0,
                    /*tile*/ TD, C_IN, BT,
                    /*tensor*/ 1u << 30, 1u << 30, 1u << 30,
                    /*stride0 (C step)*/ (u64)D_TOT,
                    /*stride1 (b step)*/ (u64)C_IN * D_TOT);
        __builtin_amdgcn_s_wait_tensorcnt(0);
    }
    __syncthreads();

    // ---- preload v[b, c, d] for this wave's (M rows x 4 d-subtiles) ----
    float vreg[8][4];
    if (MODE != 0) {
        for (int i = 0; i < 8; ++i) {
            const int m = i + half * 8;
            for (int sub = 0; sub < 4; ++sub) {
                const int d = sub * 16 + l16;
                vreg[i][sub] = v[((size_t)(b0 + m) * C_OUT + c) * D_OUT + d];
            }
        }
    }

    v8f s_reg[4];
    for (int sub = 0; sub < 4; ++sub) s_reg[sub] = (v8f)(0.0f);

    const size_t wbase0 = (size_t)c * C_IN * W_CSTRIDE;     // w[c,0,0,0]
    const u32 woff[2] = {
        (u32)(uintptr_t)&wl[(size_t)(c * 2 + 0) * WCHUNK],
        (u32)(uintptr_t)&wl[(size_t)(c * 2 + 1) * WCHUNK] };

    // ---- TDM: issue first w chunk: w[c, 0:8, D0:D0+TDC, 0:64] ----
    tdm_load_3d(woff[0], w + wbase0 + (size_t)D0 * D_OUT,
                /*tile*/ D_OUT, TDC, C_IN,
                /*tensor*/ 1u << 30, 1u << 30, 1u << 30,
                /*stride0 (D step)*/ (u64)D_OUT,
                /*stride1 (C step)*/ (u64)W_CSTRIDE);

    for (int ch = 0; ch < NCH; ++ch) {
        const int p = ch & 1;
        if (ch + 1 < NCH) {
            // buffer p^1 was consumed in chunk ch-1; drain DS before TDM
            // overwrites it (TENSOR ops are unordered vs DS).
            asm volatile("s_wait_dscnt 0x0" ::: "memory");
            tdm_load_3d(woff[p ^ 1],
                        w + wbase0 + (size_t)(D0 + (ch + 1) * TDC) * D_OUT,
                        D_OUT, TDC, C_IN,
                        1u << 30, 1u << 30, 1u << 30,
                        (u64)D_OUT, (u64)W_CSTRIDE);
            __builtin_amdgcn_s_wait_tensorcnt(1);   // chunk ch landed
        } else {
            __builtin_amdgcn_s_wait_tensorcnt(0);
        }
        asm volatile("" ::: "memory");              // LDS reload barrier

        const float* wt = &wl[(size_t)(c * 2 + p) * WCHUNK]; // [C][dd2][d]

        for (int dd2 = 0; dd2 < TDC; ++dd2) {
            const int dd = ch * TDC + dd2;
            const int Dv = D0 + dd;

            // A matrix (x): 16x4 fp32 layout, lane half selects K pair
            v2f a0, a1;
            a0.x = xs[(l16 * C_IN + (2 * half + 0)) * TD + dd];
            a0.y = xs[(l16 * C_IN + (2 * half + 1)) * TD + dd];
            a1.x = xs[(l16 * C_IN + 4 + (2 * half + 0)) * TD + dd];
            a1.y = xs[(l16 * C_IN + 4 + (2 * half + 1)) * TD + dd];

            // u tile: 4 subtiles x (2 chained K-steps of WMMA), B from LDS
            v8f u[4];
            for (int sub = 0; sub < 4; ++sub) {
                const int d = sub * 16 + l16;
                v2f bm0, bm1;
                bm0.x = wt[((2 * half + 0) * TDC + dd2) * D_OUT + d];
                bm0.y = wt[((2 * half + 1) * TDC + dd2) * D_OUT + d];
                bm1.x = wt[((4 + 2 * half + 0) * TDC + dd2) * D_OUT + d];
                bm1.y = wt[((4 + 2 * half + 1) * TDC + dd2) * D_OUT + d];
                v8f zero = (v8f)(0.0f);
                v8f acc = __builtin_amdgcn_wmma_f32_16x16x4_f32(
                    false, a0, false, bm0, (short)0, zero, false, false);
                u[sub] = __builtin_amdgcn_wmma_f32_16x16x4_f32(
                    false, a1, false, bm1, (short)0, acc, false, false);
            }

            if (MODE != 0) {
                // delta-b = sum_c u*v  (cross-wave reduce via ds_add_f32)
                for (int idx = tid; idx < BT * D_OUT; idx += BLK) bsum[idx] = 0.0f;
                __syncthreads();
                for (int sub = 0; sub < 4; ++sub) {
                    const int d = sub * 16 + l16;
                    for (int i = 0; i < 8; ++i) {
                        const int m = i + half * 8;
                        atomicAdd(&bsum[m * D_OUT + d], u[sub][i] * vreg[i][sub]);
                    }
                }
                __syncthreads();

                // per-row softmax over d (row m handled by wave m)
                {
                    const int m = wave;
                    const size_t bIdx = ((size_t)(b0 + m) * D_TOT + Dv) * D_OUT;
                    float bo0 = 0.0f, bo1 = 0.0f;
                    if (MODE == 2) {
                        bo0 = bglob[bIdx + lane];
                        bo1 = bglob[bIdx + lane + 32];
                    }
                    float t0 = bo0 + bsum[m * D_OUT + lane];
                    float t1 = bo1 + bsum[m * D_OUT + lane + 32];
                    if (MODE == 1) {
                        bglob[bIdx + lane]      = t0;
                        bglob[bIdx + lane + 32] = t1;
                    }
                    float mx = fmaxf(t0, t1);
                    for (int off = 16; off > 0; off >>= 1)
                        mx = fmaxf(mx, __shfl_xor(mx, off));
                    const float e0 = __expf(t0 - mx);
                    const float e1 = __expf(t1 - mx);
                    float sm = e0 + e1;
                    for (int off = 16; off > 0; off >>= 1)
                        sm += __shfl_xor(sm, off);
                    const float inv = 1.0f / sm;
                    ccs[m * D_OUT + lane]      = e0 * inv;
                    ccs[m * D_OUT + lane + 32] = e1 * inv;
                }
                __syncthreads();

                // s += cc * u
                for (int sub = 0; sub < 4; ++sub) {
                    const int d = sub * 16 + l16;
                    for (int i = 0; i < 8; ++i) {
                        const int m = i + half * 8;
                        s_reg[sub][i] += ccs[m * D_OUT + d] * u[sub][i];
                    }
                }
                __syncthreads();
            } else {
                const float cci = 1.0f / 64.0f;   // softmax of zeros over 64
                for (int sub = 0; sub < 4; ++sub)
                    for (int i = 0; i < 8; ++i)
                        s_reg[sub][i] += cci * u[sub][i];
            }
        }
    }

    // ---- reduce this D-tile's partial s into global accumulator ----
    for (int sub = 0; sub < 4; ++sub) {
        const int d = sub * 16 + l16;
        for (int i = 0; i < 8; ++i) {
            const int m = i + half * 8;
            atomicAdd(&sacc[((size_t)(b0 + m) * C_OUT + c) * D_OUT + d],
                      s_reg[sub][i]);
        }
    }
}

// v = squash(s) over channel dim c; also zero s accumulator for the next pass.
__global__ __launch_bounds__(64) void squash_kernel(float* __restrict__ sacc,
                                                    float* __restrict__ v)
{
    const int b = blockIdx.x;     // 64
    const int d = threadIdx.x;    // 64
    float sv[C_OUT];
    float n2 = 0.0f;
    for (int c = 0; c < C_OUT; ++c) {
        sv[c] = sacc[((size_t)b * C_OUT + c) * D_OUT + d];
        n2 += sv[c] * sv[c];
    }
    const float n = sqrtf(n2);
    const float f = (n2 / (1.0f + n2)) / n;   // EPS = 0 per reference
    for (int c = 0; c < C_OUT; ++c) {
        const size_t idx = ((size_t)b * C_OUT + c) * D_OUT + d;
        v[idx]    = f * sv[c];
        sacc[idx] = 0.0f;
    }
}

extern "C" void kernel_launch(void* const* d_in, const int* in_sizes, int n_in,
                              void* d_out, int out_size, void* d_ws, size_t ws_size,
                              hipStream_t stream)
{
    (void)in_sizes; (void)n_in; (void)ws_size;
    const float* x = (const float*)d_in[0];   // (64, 8, 1152)
    const float* w = (const float*)d_in[1];   // (16, 8, 1152, 64)
    float* out = (float*)d_out;               // (64, 16, 64) = s of iter 3

    const size_t SV = (size_t)B_TOT * C_OUT * D_OUT;          // 65536 floats
    float* s_ws = (float*)d_ws;
    float* v_ws = s_ws + SV;
    float* b_ws = v_ws + SV;                                  // (64,1152,64)

    dim3 grid(D_TOT / TD, B_TOT / BT);   // 36 x 4 workgroups
    dim3 blk(BLK);                       // 512 threads = 16 wave32
    const size_t dynlds = (size_t)2 * NW * WCHUNK * sizeof(float);  // 256 KB

    (void)hipMemsetAsync(s_ws, 0, SV * sizeof(float), stream);

    // iter 1: cc uniform, accumulate s1
    routing_pass<0><<<grid, blk, dynlds, stream>>>(x, w, nullptr, nullptr, s_ws);
    squash_kernel<<<B_TOT, D_OUT, 0, stream>>>(s_ws, v_ws);   // v1, zero s

    // iter 1->2 fused: b = sum_c u*v1 (stored), cc = softmax(b), s2 += cc*u
    routing_pass<1><<<grid, blk, dynlds, stream>>>(x, w, v_ws, b_ws, s_ws);
    squash_kernel<<<B_TOT, D_OUT, 0, stream>>>(s_ws, v_ws);   // v2, zero s

    // iter 2->3 fused: b += sum_c u*v2, cc = softmax(b), s3 += cc*u -> d_out
    (void)hipMemsetAsync(out, 0, (size_t)out_size * sizeof(float), stream);
    routing_pass<2><<<grid, blk, dynlds, stream>>>(x, w, v_ws, b_ws, out);
}